// MemristorConv1d_50826642981442
// MI455X (gfx1250) — compile-verified
//
#include <hip/hip_runtime.h>
#include <cstdint>

#define B_    4
#define F_    512
#define T_    1000
#define K_    31
#define PAD_  15
#define DLEN  1040      // reads reach index 1038 = 992+15+28+3
#define NTILE 63        // ceil(1000/16)

typedef float v2f __attribute__((ext_vector_type(2)));
typedef float v8f __attribute__((ext_vector_type(8)));

// dac(x*INPUT_FACTOR) followed by v*(poly_hrs(v^2) - poly_lrs(v^2)); matches the
// reference op order (clip -> rne-round*127 -> /127 -> *0.6).
__device__ __forceinline__ float dac_g(float x) {
  float v = x * 0.25f;
  v = fminf(1.0f, fmaxf(-1.0f, v));
  v = (rintf(v * 127.0f) / 127.0f) * 0.6f;
  float v2 = v * v;
  float ph = 2e-06f + v2 * (5e-08f + v2 * 1e-09f);
  float pl = 3e-04f + v2 * (4e-06f + v2 * 2e-07f);
  return v * (ph - pl);
}

// adc: clip(rne_round(i*5000*256)/256, -16, 16); *256 and /256 are exact.
__device__ __forceinline__ float adc_q(float x) {
  float r = rintf((x * 5000.0f) * 256.0f) * 0.00390625f;
  return fminf(16.0f, fmaxf(-16.0f, r));
}

__global__ __launch_bounds__(256)
void memristor_conv1d_kernel(const float* __restrict__ x,
                             const float* __restrict__ rpos,
                             const float* __restrict__ rneg,
                             const float* __restrict__ bias,
                             float* __restrict__ out) {
  __shared__ float dls[DLEN];        // g(dac(x)) with 15 zeros of halo each side
  __shared__ float wraw[2 * 3 * K_]; // raw r_pos rows (93) then r_neg rows (93)
  __shared__ float w2[3 * 32];       // (r_pos - r_neg), tap-padded to 32 (w[31]=0)

  const int tid = threadIdx.x;
  const int f   = blockIdx.x % F_;
  const int b   = blockIdx.x / F_;

  // --- stage the 6 filter rows for this feature via async global->LDS ---
  if (tid < 186) {
    const float* src;
    if (tid < 93) src = rpos + (size_t)(tid / 31) * F_ * K_ + (size_t)f * K_ + (tid % 31);
    else {
      int j = tid - 93;
      src = rneg + (size_t)(j / 31) * F_ * K_ + (size_t)f * K_ + (j % 31);
    }
    uint32_t loff = (uint32_t)(uintptr_t)(&wraw[tid]);  // low 32 bits = LDS offset
    asm volatile("global_load_async_to_lds_b32 %0, %1, off"
                 :: "v"(loff), "v"(src) : "memory");
  }

  // --- pointwise front-end: fill d array (zeros in the halo) ---
  const float* xin = x + ((size_t)b * F_ + f) * T_;
  for (int i = tid; i < DLEN; i += 256) {
    int t = i - PAD_;
    float dv = 0.0f;
    if (t >= 0 && t < T_) dv = dac_g(xin[t]);
    dls[i] = dv;
  }

  asm volatile("s_wait_asynccnt 0" ::: "memory");
  __syncthreads();

  if (tid < 96) {
    int p = tid >> 5, k = tid & 31;
    float w = 0.0f;
    if (k < K_) w = wraw[p * K_ + k] - wraw[93 + p * K_ + k];
    w2[p * 32 + k] = w;
  }
  __syncthreads();

  const int wave = tid >> 5;
  const int lane = tid & 31;
  const int half = lane >> 4;   // wave32 halves
  const int lm   = lane & 15;

  // B matrix (4x16 f32) per q-group: column n = lm holds filter p=n (n<3),
  // VGPR0 = tap K0/K2 by half, VGPR1 = tap K1/K3 (mirrors C/D row striping).
  float b0[8], b1[8];
#pragma unroll
  for (int q = 0; q < 8; ++q) {
    float u0 = 0.0f, u1 = 0.0f;
    if (lm < 3) {
      u0 = w2[lm * 32 + 4 * q + 2 * half];
      u1 = w2[lm * 32 + 4 * q + 2 * half + 1];
    }
    b0[q] = u0; b1[q] = u1;
  }

  const float bw = (lm == 0) ? 4.0f : (lm == 1) ? 2.0f : (lm == 2) ? 1.0f : 0.0f;
  const float bf = bias[f];
  float* op = out + ((size_t)b * F_ + f) * T_;

  for (int tile = wave; tile < NTILE; tile += 8) {
    const int t0 = tile * 16;
    v8f c = {0.f, 0.f, 0.f, 0.f, 0.f, 0.f, 0.f, 0.f};
#pragma unroll
    for (int q = 0; q < 8; ++q) {
      // A (16x4 f32): lane<16 -> row m=lm taps {4q+0,4q+1}; lane>=16 -> {4q+2,4q+3}
      int ai = t0 + lm + 4 * q + 2 * half;
      v2f a;  a.x  = dls[ai];   a.y  = dls[ai + 1];
      v2f bb; bb.x = b0[q];     bb.y = b1[q];
      c = __builtin_amdgcn_wmma_f32_16x16x4_f32(
              /*neg_a=*/false, a, /*neg_b=*/false, bb,
              /*c_mod=*/(short)0, c, /*reuse_a=*/false, /*reuse_b=*/false);
    }

    // D[m,n]: VGPR r = row m=r+8*half, lane%16 = n. ADC per column, weight by
    // {4,2,1}, sum columns 0..2 (cols >=3 are exactly 0 -> adc->0).
    float res[8];
#pragma unroll
    for (int r = 0; r < 8; ++r) {
      float z = adc_q(c[r]) * bw;
      z += __shfl_down(z, 1);
      z += __shfl_down(z, 2);
      res[r] = z * 0.02f + bf;   // OUTPUT_FACTOR + bias
    }

    if (lm == 0) {               // lanes 0 and 16 hold rows 0..7 / 8..15
      int tb = t0 + half * 8;
      if (tb + 7 < T_) {
        *(float4*)(op + tb)     = make_float4(res[0], res[1], res[2], res[3]);
        *(float4*)(op + tb + 4) = make_float4(res[4], res[5], res[6], res[7]);
      } else {
#pragma unroll
        for (int j = 0; j < 8; ++j)
          if (tb + j < T_) op[tb + j] = res[j];
      }
    }
  }
}

extern "C" void kernel_launch(void* const* d_in, const int* in_sizes, int n_in,
                              void* d_out, int out_size, void* d_ws, size_t ws_size,
                              hipStream_t stream) {
  (void)in_sizes; (void)n_in; (void)d_ws; (void)ws_size; (void)out_size;
  const float* x    = (const float*)d_in[0];
  const float* rpos = (const float*)d_in[1];
  const float* rneg = (const float*)d_in[2];
  const float* bias = (const float*)d_in[3];
  float* out        = (float*)d_out;
  memristor_conv1d_kernel<<<dim3(B_ * F_), dim3(256), 0, stream>>>(x, rpos, rneg, bias, out);
}